// HybridMemory_62079457296450
// MI455X (gfx1250) — compile-verified
//
#include <hip/hip_runtime.h>
#include <math.h>

// ---------------------------------------------------------------------------
// Problem constants (fixed by the reference setup)
// ---------------------------------------------------------------------------
#define B_SZ 256      // batch
#define F_SZ 256      // feature dim
#define M_SZ 65536    // memory bank rows
#define C_SZ 8192     // classes
#define TEMP_F 0.05f
#define EPS_F  1e-6f

typedef __attribute__((ext_vector_type(2))) float v2f;
typedef __attribute__((ext_vector_type(8))) float v8f;

#define WMMA_F32(a, b, c) \
  __builtin_amdgcn_wmma_f32_16x16x4_f32(false, (a), false, (b), (short)0, (c), false, false)

// ---------------------------------------------------------------------------
// Block reduction (wave32-aware: 256 threads = 8 waves on CDNA5)
// ---------------------------------------------------------------------------
__device__ __forceinline__ float blockReduceSum(float v, float* sm) {
#pragma unroll
  for (int o = 16; o > 0; o >>= 1) v += __shfl_xor(v, o, 32);
  const int wave = threadIdx.x >> 5;
  const int lane = threadIdx.x & 31;
  if (lane == 0) sm[wave] = v;
  __syncthreads();
  const int nw = blockDim.x >> 5;
  if (wave == 0) {
    float r = (lane < nw) ? sm[lane] : 0.0f;
#pragma unroll
    for (int o = 4; o > 0; o >>= 1) r += __shfl_xor(r, o, 32);
    if (lane == 0) sm[0] = r;
  }
  __syncthreads();
  float r = sm[0];
  __syncthreads();
  return r;
}

// ---------------------------------------------------------------------------
// Zero a float region
// ---------------------------------------------------------------------------
__global__ void zero_kernel(float* __restrict__ p, int n) {
  int i = blockIdx.x * blockDim.x + threadIdx.x;
  if (i < n) p[i] = 0.0f;
}

// ---------------------------------------------------------------------------
// Row-wise L2 normalization of the three [B,F] inputs (one block per row)
// ---------------------------------------------------------------------------
__global__ void normalize3_kernel(const float* __restrict__ f0, const float* __restrict__ f1,
                                  const float* __restrict__ f2,
                                  float* __restrict__ o0, float* __restrict__ o1,
                                  float* __restrict__ o2) {
  __shared__ float sm[8];
  const int r = blockIdx.x;
  const int t = threadIdx.x;
  const float* src[3] = {f0, f1, f2};
  float* dst[3] = {o0, o1, o2};
#pragma unroll
  for (int i = 0; i < 3; ++i) {
    const float x = src[i][r * F_SZ + t];
    const float s = blockReduceSum(x * x, sm);
    dst[i][r * F_SZ + t] = x * (1.0f / sqrtf(s));
  }
}

// ---------------------------------------------------------------------------
// Segment-sum of memory-bank feature rows into per-class sums + class counts.
// (linearity: segment_sum of dot == dot with segment-summed features)
// ---------------------------------------------------------------------------
__global__ void class_scatter_kernel(const float* __restrict__ features,
                                     const int* __restrict__ labels,
                                     float* __restrict__ classSum,
                                     float* __restrict__ classCount) {
  const int m = blockIdx.x;
  const int c = labels[m];
  const float* src = features + (size_t)m * F_SZ;
  float* dst = classSum + (size_t)c * F_SZ;
  unsafeAtomicAdd(&dst[threadIdx.x], src[threadIdx.x]);
  if (threadIdx.x == 0) unsafeAtomicAdd(&classCount[c], 1.0f);
}

// ---------------------------------------------------------------------------
// fp32 WMMA GEMM:  C[M,N] = (A[M,K] @ B[N,K]^T) * colScale
//   colScale = scale                         if counts == nullptr
//   colScale = scale / max(counts[n], 1)     otherwise
//
// Each wave computes a 16x32 output strip (2 accumulators sharing one A
// fragment per K-step). 2 accumulators (16 VGPRs) keeps register pressure
// low enough that the pre-RA scheduler clusters the independent b64 loads
// of the unrolled K-window instead of serializing load->wait0->wmma (the
// failure mode observed with 4 accumulators). __launch_bounds__(256,1)
// relaxes the occupancy-driven VGPR budget for the same reason.
// Addressing: uniform SGPR base + 32-bit lane offset (saddr+voffset form).
// EXEC all-ones (full blocks, no divergent exits) as WMMA requires.
// A-layout per ISA: lane<16 -> K=0,1 ; lane>=16 -> K=2,3 ; row = lane&15.
// ---------------------------------------------------------------------------
__global__ void __launch_bounds__(256, 1)
wmma_gemm_nt_kernel(const float* __restrict__ A,
                    const float* __restrict__ B,
                    float* __restrict__ C,
                    const float* __restrict__ counts,
                    int M, int N, int K, float scale) {
  const int wave = threadIdx.x >> 5;
  const int lane = threadIdx.x & 31;
  const int task = blockIdx.x * (blockDim.x >> 5) + wave;
  const int ngrp = N >> 5;                        // groups of 32 columns
  const int m0 = (task / ngrp) << 4;
  const int n0 = (task % ngrp) << 5;

  const int r = lane & 15;
  const int kh = (lane >> 4) << 1;                // K sub-offset: 0 or 2
  const int aBase  = (m0 + r) * K + kh;           // 32-bit lane offsets
  const int b0Base = (n0 + r) * K + kh;
  const int b1Base = b0Base + 16 * K;

  v8f acc0 = {}, acc1 = {};
#pragma unroll 4
  for (int k = 0; k < K; k += 4) {
    const v2f a  = *(const v2f*)(A + aBase + k);
    const v2f b0 = *(const v2f*)(B + b0Base + k);
    const v2f b1 = *(const v2f*)(B + b1Base + k);
    acc0 = WMMA_F32(a, b0, acc0);
    acc1 = WMMA_F32(a, b1, acc1);
  }

  const int mbase = m0 + ((lane >> 4) << 3);      // rows 0-7 (lanes<16) or 8-15
  const v8f accs[2] = {acc0, acc1};
#pragma unroll
  for (int t = 0; t < 2; ++t) {
    const int n = n0 + (t << 4) + (lane & 15);
    float s = scale;
    if (counts) {
      const float c = counts[n];
      s = scale / (c > 0.0f ? c : 1.0f);
    }
#pragma unroll
    for (int v = 0; v < 8; ++v) {
      C[(size_t)(mbase + v) * N + n] = accs[t][v] * s;
    }
  }
}

// ---------------------------------------------------------------------------
// spcl loss per sample: masked softmax over 8192 classes, pick target class.
// ---------------------------------------------------------------------------
__global__ void spcl_kernel(const float* __restrict__ sim_agg,   // [B, C]
                            const float* __restrict__ classCount,
                            const int* __restrict__ labels,
                            const int* __restrict__ indexes,
                            float* __restrict__ spcl_b) {
  __shared__ float sm[8];
  const int b = blockIdx.x;
  const int target = labels[indexes[b]];
  const float* row = sim_agg + (size_t)b * C_SZ;
  float sum = 0.0f, tval = 0.0f;
  for (int c = threadIdx.x; c < C_SZ; c += blockDim.x) {
    const float mask = (classCount[c] > 0.0f) ? 1.0f : 0.0f;
    const float e = expf(row[c]) * mask;
    sum += e;
    if (c == target) tval = e;
  }
  sum = blockReduceSum(sum, sm);
  tval = blockReduceSum(tval, sm);
  if (threadIdx.x == 0) spcl_b[b] = -logf(tval / (sum + EPS_F) + EPS_F);
}

// ---------------------------------------------------------------------------
// co / ad losses: one block per anchor b, thread j = pair column.
// ---------------------------------------------------------------------------
__global__ void pair_loss_kernel(const float* __restrict__ simo,
                                 const float* __restrict__ simg,
                                 const float* __restrict__ sima,
                                 const int* __restrict__ labels,
                                 const int* __restrict__ indexes,
                                 float* __restrict__ co_b, float* __restrict__ ad_b) {
  __shared__ float sm[8];
  const int b = blockIdx.x;
  const int j = threadIdx.x;
  const int tb = labels[indexes[b]];
  const int tj = labels[indexes[j]];
  const float eo = expf(simo[b * B_SZ + j]);
  const float eg = expf(simg[b * B_SZ + j]);
  const float ea = expf(sima[b * B_SZ + j]);
  const float same = (tb == tj) ? 1.0f : 0.0f;

  const float sum_o = blockReduceSum(eo, sm);
  const float sum_g = blockReduceSum(eg, sm);
  const float sum_a = blockReduceSum(ea, sm);
  const float s_oa  = blockReduceSum(same * (eo + ea), sm);
  const float s_g   = blockReduceSum(same * eg, sm);

  // c_co = sum_j (eo+ea)*diff + sum_j eg ; c_ad = sum_j eg*diff + sum_a + sum_o
  const float c_co = (sum_o + sum_a - s_oa) + sum_g;
  const float c_ad = (sum_g - s_g) + sum_a + sum_o;

  const float term_o = -logf(eo / (eo + c_co + EPS_F) + EPS_F);
  const float term_a = -logf(ea / (ea + c_co + EPS_F) + EPS_F);
  const float term_g = -logf(eg / (eg + c_ad + EPS_F) + EPS_F);

  const float co = blockReduceSum(same * (term_o + term_a), sm);
  const float ad = blockReduceSum(same * term_g, sm);
  if (j == 0) { co_b[b] = co; ad_b[b] = ad; }
}

// ---------------------------------------------------------------------------
// Deterministic final means -> out = {spcl, ad, co}
// ---------------------------------------------------------------------------
__global__ void finish_kernel(const float* __restrict__ spcl_b,
                              const float* __restrict__ co_b,
                              const float* __restrict__ ad_b,
                              float* __restrict__ out) {
  __shared__ float sm[8];
  const int t = threadIdx.x;
  const float s = blockReduceSum(spcl_b[t], sm);
  const float a = blockReduceSum(ad_b[t], sm);
  const float c = blockReduceSum(co_b[t], sm);
  if (t == 0) {
    out[0] = s * (1.0f / B_SZ);
    out[1] = a * (1.0f / B_SZ);
    out[2] = c * (1.0f / B_SZ);
  }
}

// ---------------------------------------------------------------------------
// Host launcher
// ---------------------------------------------------------------------------
extern "C" void kernel_launch(void* const* d_in, const int* in_sizes, int n_in,
                              void* d_out, int out_size, void* d_ws, size_t ws_size,
                              hipStream_t stream) {
  (void)in_sizes; (void)n_in; (void)out_size; (void)ws_size;

  const float* feat      = (const float*)d_in[0];
  const float* feat_gen  = (const float*)d_in[1];
  const float* feat_aug  = (const float*)d_in[2];
  const float* features  = (const float*)d_in[3];
  const int*   indexes   = (const int*)d_in[4];
  const int*   labels    = (const int*)d_in[5];
  float* out = (float*)d_out;

  // Workspace layout (floats)
  float* ws = (float*)d_ws;
  float* ori_n      = ws;                     ws += B_SZ * F_SZ;
  float* gen_n      = ws;                     ws += B_SZ * F_SZ;
  float* aug_n      = ws;                     ws += B_SZ * F_SZ;
  float* sim_o      = ws;                     ws += B_SZ * B_SZ;
  float* sim_g      = ws;                     ws += B_SZ * B_SZ;
  float* sim_a      = ws;                     ws += B_SZ * B_SZ;
  float* classSum   = ws;                     ws += (size_t)C_SZ * F_SZ;
  float* classCount = ws;                     ws += C_SZ;
  float* sim_agg    = ws;                     ws += (size_t)B_SZ * C_SZ;
  float* spcl_b     = ws;                     ws += B_SZ;
  float* co_b       = ws;                     ws += B_SZ;
  float* ad_b       = ws;                     ws += B_SZ;

  const float invT = 1.0f / TEMP_F;

  // 1) zero class accumulators (classSum + classCount are contiguous)
  {
    const int n = C_SZ * F_SZ + C_SZ;
    zero_kernel<<<(n + 255) / 256, 256, 0, stream>>>(classSum, n);
  }

  // 2) normalize the three feature matrices
  normalize3_kernel<<<B_SZ, F_SZ, 0, stream>>>(feat, feat_gen, feat_aug,
                                               ori_n, gen_n, aug_n);

  // 3) scatter memory-bank rows into per-class sums (+ counts)
  class_scatter_kernel<<<M_SZ, F_SZ, 0, stream>>>(features, labels, classSum, classCount);

  // 4) small pairwise similarity GEMMs (fp32 WMMA), scaled by 1/TEMP
  {
    const int tasks = (B_SZ / 16) * (B_SZ / 32);   // 128 wave-tasks (16x32 strips)
    const int blocks = tasks / 8;                  // 8 waves per block
    wmma_gemm_nt_kernel<<<blocks, 256, 0, stream>>>(ori_n, ori_n, sim_o, nullptr,
                                                    B_SZ, B_SZ, F_SZ, invT);
    wmma_gemm_nt_kernel<<<blocks, 256, 0, stream>>>(ori_n, gen_n, sim_g, nullptr,
                                                    B_SZ, B_SZ, F_SZ, invT);
    wmma_gemm_nt_kernel<<<blocks, 256, 0, stream>>>(ori_n, aug_n, sim_a, nullptr,
                                                    B_SZ, B_SZ, F_SZ, invT);
  }

  // 5) class-aggregated similarity GEMM: aug_n @ classSum^T, per-class scaling
  {
    const int tasks = (B_SZ / 16) * (C_SZ / 32);   // 4096 wave-tasks
    const int blocks = tasks / 8;                  // 512 blocks
    wmma_gemm_nt_kernel<<<blocks, 256, 0, stream>>>(aug_n, classSum, sim_agg, classCount,
                                                    B_SZ, C_SZ, F_SZ, invT);
  }

  // 6) losses
  spcl_kernel<<<B_SZ, 256, 0, stream>>>(sim_agg, classCount, labels, indexes, spcl_b);
  pair_loss_kernel<<<B_SZ, B_SZ, 0, stream>>>(sim_o, sim_g, sim_a, labels, indexes,
                                              co_b, ad_b);
  finish_kernel<<<1, B_SZ, 0, stream>>>(spcl_b, co_b, ad_b, out);
}